// Group_60017872994321
// MI455X (gfx1250) — compile-verified
//
#include <hip/hip_runtime.h>
#include <hip/hip_bf16.h>

typedef float v2f __attribute__((ext_vector_type(2)));
typedef float v8f __attribute__((ext_vector_type(8)));

#define BATCH 32
#define NPTS  8192
#define NGRP  512
#define KNN   32
#define BIGD  1e10f

// ---------------------------------------------------------------------------
// Kernel 1: farthest point sampling. One 256-thread block per batch.
// min_d lives in LDS (32KB), point coords cached in registers (32/thread).
// 511 sequential steps: update running min, block-wide argmax (first-max tie
// rule like jnp.argmax), emit center coords into the center region of d_out.
// ---------------------------------------------------------------------------
__global__ __launch_bounds__(256) void fps_kernel(const float* __restrict__ xyz,
                                                  float* __restrict__ centers) {
  const int b = blockIdx.x;
  const int t = threadIdx.x;

  __shared__ float mind[NPTS];     // 32 KB
  __shared__ float red_d[256];
  __shared__ int   red_i[256];

  const float* base = xyz + (size_t)b * NPTS * 3;

  float px[32], py[32], pz[32];
#pragma unroll
  for (int r = 0; r < 32; ++r) {
    int i = t + (r << 8);
    const float* p = base + (size_t)i * 3;
    px[r] = p[0]; py[r] = p[1]; pz[r] = p[2];
    mind[i] = BIGD;
  }
  if (t == 0) {  // idx[0] == 0 always
    float* c = centers + (size_t)b * NGRP * 3;
    c[0] = base[0]; c[1] = base[1]; c[2] = base[2];
  }
  __syncthreads();

  int last = 0;
  for (int s = 1; s < NGRP; ++s) {
    const float lx = base[(size_t)last * 3 + 0];
    const float ly = base[(size_t)last * 3 + 1];
    const float lz = base[(size_t)last * 3 + 2];

    float bestd = -1.0f;
    int   besti = 0;
#pragma unroll
    for (int r = 0; r < 32; ++r) {
      int i = t + (r << 8);
      float dx = px[r] - lx, dy = py[r] - ly, dz = pz[r] - lz;
      float d = dx * dx + dy * dy + dz * dz;
      float m = fminf(mind[i], d);
      mind[i] = m;
      if (m > bestd) { bestd = m; besti = i; }   // strict > keeps first max
    }
    red_d[t] = bestd; red_i[t] = besti;
    __syncthreads();
    for (int off = 128; off > 0; off >>= 1) {
      if (t < off) {
        float od = red_d[t + off]; int oi = red_i[t + off];
        if (od > red_d[t] || (od == red_d[t] && oi < red_i[t])) {
          red_d[t] = od; red_i[t] = oi;
        }
      }
      __syncthreads();
    }
    last = red_i[0];
    if (t == 0) {
      float* c = centers + ((size_t)b * NGRP + s) * 3;
      c[0] = base[(size_t)last * 3 + 0];
      c[1] = base[(size_t)last * 3 + 1];
      c[2] = base[(size_t)last * 3 + 2];
    }
    __syncthreads();  // protect red_* WAR for next iteration
  }
}

// ---------------------------------------------------------------------------
// Kernel 2: fused KNN + gather. One wave (32 threads) per 16 centers.
// d2 tile = single V_WMMA_F32_16X16X4_F32:
//   A[m][k] = (cx, cy, cz, |c|^2)    (16x4)
//   B[k][n] = (-2px, -2py, -2pz, 1)  (4x16)
//   C[m][n] = |p_n|^2 (broadcast)
//   D       = |c|^2 - 2 c.p + |p|^2  = exact fp32 squared distance
// Each lane keeps a register-resident sorted top-32 over its half of the
// point stream; two halves merged at the end (stable ascending, ties by idx).
// ---------------------------------------------------------------------------
__global__ __launch_bounds__(32) void knn_group_kernel(const float* __restrict__ xyz,
                                                       const float* __restrict__ centers,
                                                       float* __restrict__ out_nb) {
  const int b    = blockIdx.y;
  const int g0   = blockIdx.x << 4;   // first of 16 centers
  const int lane = threadIdx.x;       // wave32
  const int n16  = lane & 15;
  const int half = lane >> 4;         // 0 or 1

  __shared__ float4 pbuf[32];
  __shared__ float  d2buf[16][32];
  __shared__ float  mg_d[2][16][32];
  __shared__ int    mg_i[2][16][32];

  const float* base = xyz + (size_t)b * NPTS * 3;

  // A matrix: ISA layout -> lanes 0-15 hold (K0,K1), lanes 16-31 hold (K2,K3)
  const float* cp = centers + ((size_t)b * NGRP + g0 + n16) * 3;
  float cx = cp[0], cy = cp[1], cz = cp[2];
  float cn2 = cx * cx + cy * cy + cz * cz;
  v2f A;
  A.x = half ? cz  : cx;
  A.y = half ? cn2 : cy;

  float ld[32];
  int   li[32];
#pragma unroll
  for (int j = 0; j < 32; ++j) { ld[j] = 3.0e38f; li[j] = 0x7fffffff; }

  for (int p0 = 0; p0 < NPTS; p0 += 32) {
    {  // stage 32 points (x, y, z, |p|^2) in LDS
      const float* p = base + (size_t)(p0 + lane) * 3;
      float x = p[0], y = p[1], z = p[2];
      pbuf[lane] = make_float4(x, y, z, x * x + y * y + z * z);
    }
    __syncthreads();

    float4 q0 = pbuf[n16];
    float4 q1 = pbuf[16 + n16];
    v2f B0, B1;
    B0.x = half ? (-2.0f * q0.z) : (-2.0f * q0.x);
    B0.y = half ? 1.0f           : (-2.0f * q0.y);
    B1.x = half ? (-2.0f * q1.z) : (-2.0f * q1.x);
    B1.y = half ? 1.0f           : (-2.0f * q1.y);
    v8f C0, C1;
#pragma unroll
    for (int v = 0; v < 8; ++v) { C0[v] = q0.w; C1[v] = q1.w; }

    // (neg_a, A, neg_b, B, c_mod, C, reuse_a, reuse_b)
    v8f D0 = __builtin_amdgcn_wmma_f32_16x16x4_f32(false, A, false, B0, (short)0, C0, false, false);
    v8f D1 = __builtin_amdgcn_wmma_f32_16x16x4_f32(false, A, false, B1, (short)0, C1, false, false);

#pragma unroll
    for (int v = 0; v < 8; ++v) {  // D layout: M = v + 8*half, N = n16
      int m = v + (half << 3);
      d2buf[m][n16]      = D0[v];
      d2buf[m][16 + n16] = D1[v];
    }
    __syncthreads();

    // lane handles center row n16, columns [half*16, half*16+16)
#pragma unroll
    for (int j = 0; j < 16; ++j) {
      int   col = (half << 4) + j;
      float d   = d2buf[n16][col];
      int   gi  = p0 + col;
      if (d < ld[31]) {  // strict <: equal dists keep earlier index
        ld[31] = d; li[31] = gi;
#pragma unroll
        for (int q = 31; q > 0; --q) {
          float a = ld[q - 1], c = ld[q];
          int   xa = li[q - 1], xc = li[q];
          bool sw = c < a;
          ld[q - 1] = sw ? c : a;   ld[q] = sw ? a : c;
          li[q - 1] = sw ? xc : xa; li[q] = sw ? xa : xc;
        }
      }
    }
    __syncthreads();
  }

  // publish both half-lists, then merge per center (stable, ties -> lower idx)
#pragma unroll
  for (int j = 0; j < 32; ++j) { mg_d[half][n16][j] = ld[j]; mg_i[half][n16][j] = li[j]; }
  __syncthreads();

  if (lane < 16) {
    int g = g0 + n16;
    const float* cc = centers + ((size_t)b * NGRP + g) * 3;
    float ccx = cc[0], ccy = cc[1], ccz = cc[2];
    float* ob = out_nb + ((size_t)b * NGRP + g) * KNN * 3;
    int ia = 0, ib = 0;
    for (int j = 0; j < 32; ++j) {
      float da = mg_d[0][n16][ia], db = mg_d[1][n16][ib];
      int   xa = mg_i[0][n16][ia], xb = mg_i[1][n16][ib];
      bool takeA = (da < db) || (da == db && xa < xb);
      int pi = takeA ? xa : xb;
      if (takeA) ++ia; else ++ib;
      const float* pp = base + (size_t)pi * 3;
      ob[j * 3 + 0] = pp[0] - ccx;
      ob[j * 3 + 1] = pp[1] - ccy;
      ob[j * 3 + 2] = pp[2] - ccz;
    }
  }
}

// ---------------------------------------------------------------------------
extern "C" void kernel_launch(void* const* d_in, const int* in_sizes, int n_in,
                              void* d_out, int out_size, void* d_ws, size_t ws_size,
                              hipStream_t stream) {
  (void)in_sizes; (void)n_in; (void)out_size; (void)d_ws; (void)ws_size;
  const float* xyz = (const float*)d_in[0];
  float* out = (float*)d_out;
  // d_out = neighborhood [B,G,K,3] flat, then center [B,G,3] flat
  float* centers = out + (size_t)BATCH * NGRP * KNN * 3;

  fps_kernel<<<BATCH, 256, 0, stream>>>(xyz, centers);
  knn_group_kernel<<<dim3(NGRP / 16, BATCH), 32, 0, stream>>>(xyz, centers, out);
}